// TorchQuanvLayer_56186762166802
// MI455X (gfx1250) — compile-verified
//
#include <hip/hip_runtime.h>

// Quanvolution layer for MI455X (gfx1250, wave32).
// Per 2x2 patch: 4-qubit real product state -> CNOT ring -> RY(theta) layer -> <Z_q>.
// CNOT ring + RY layer folded into one fixed 16x16 matrix M; the per-patch
// 16-dim transform is done as W = M * V (16 patches per wave) with four
// accumulating V_WMMA_F32_16X16X4_F32 ops. Memory-bound: 67 MB total traffic
// => ~2.9 us floor at 23.3 TB/s; all compute hides behind loads.

typedef float v2f __attribute__((ext_vector_type(2)));
typedef float v8f __attribute__((ext_vector_type(8)));

#define IMG_H 512
#define IMG_W 512
#define HO    256
#define WO    256
#define NB    32     // batch
#define NITER 8      // ho rows per wave

__global__ __launch_bounds__(256) void quanv_wmma_kernel(
    const float* __restrict__ x,      // [NB, 1, 512, 512]
    const float* __restrict__ theta,  // [4]
    float* __restrict__ out)          // raw [L, NB, 4] buffer (L = 256*256)
{
    const int wid  = blockIdx.x * 8 + (threadIdx.x >> 5);
    const int lane = threadIdx.x & 31;
    const int p    = lane & 15;   // patch-in-wave / WMMA row / WMMA column
    const int hi   = lane >> 4;   // lane half: selects K slice within each 16x16x4 step

    const int wo   = ((wid & 15) << 4) + p;      // 16 waves cover a 256-wide row
    const int ho0  = ((wid >> 4) & 31) << 3;     // 8 rows per wave
    const int b    = wid >> 9;                   // batch index

    // ---- Build this lane's A-matrix slices of M (loop-invariant) ----
    // M[r,j] = R[r, tau_inv(j)], tau = CNOT(0,1);CNOT(1,2);CNOT(2,3);CNOT(3,0)
    // qubit q <-> bit q of the 16-index (qubit0 = LSB, internal convention).
    float ct[4], st[4];
#pragma unroll
    for (int q = 0; q < 4; ++q) __sincosf(0.5f * theta[q], &st[q], &ct[q]);

    v2f A[4];
#pragma unroll
    for (int k = 0; k < 4; ++k) {
#pragma unroll
        for (int jj = 0; jj < 2; ++jj) {
            int j = 4 * k + 2 * hi + jj;       // column of M this slot holds
            int i = j;                          // tau^{-1}: apply involutions in order 0..3
            i ^= ((i >> 0) & 1) << 1;
            i ^= ((i >> 1) & 1) << 2;
            i ^= ((i >> 2) & 1) << 3;
            i ^= ((i >> 3) & 1) << 0;
            float m = 1.0f;
#pragma unroll
            for (int q = 0; q < 4; ++q) {
                int br = (p >> q) & 1;          // row bit (RY output index)
                int bi = (i >> q) & 1;          // col bit (RY input index)
                // RY = [[c,-s],[s,c]]
                m *= (br == bi) ? ct[q] : (br ? st[q] : -st[q]);
            }
            A[k][jj] = m;
        }
    }

    const float* xb = x + (size_t)b * (IMG_H * IMG_W);

#pragma unroll 1
    for (int it = 0; it < NITER; ++it) {
        const int ho = ho0 + it;
        // patch pixels row-major: a0=x[2ho,2wo] a1=x[2ho,2wo+1] a2=x[2ho+1,2wo] a3=x[2ho+1,2wo+1]
        const float2 ra = *(const float2*)(xb + (size_t)(2 * ho)     * IMG_W + 2 * wo);
        const float2 rb = *(const float2*)(xb + (size_t)(2 * ho + 1) * IMG_W + 2 * wo);

        float c0, s0, cq1, sq1, cq2, sq2, cq3, sq3;
        __sincosf(0.5f * ra.x, &s0,  &c0);   // qubit 0
        __sincosf(0.5f * ra.y, &sq1, &cq1);  // qubit 1
        __sincosf(0.5f * rb.x, &sq2, &cq2);  // qubit 2
        __sincosf(0.5f * rb.y, &sq3, &cq3);  // qubit 3

        // Product-state components this lane supplies in the B layout:
        // index = 4k + 2*hi + jj  (bit0=jj, bit1=hi, bit2=k&1, bit3=k>>1)
        const float f1 = hi ? sq1 : cq1;
        const float b0 = c0 * f1;            // jj = 0
        const float b1 = s0 * f1;            // jj = 1

        v8f acc = {0, 0, 0, 0, 0, 0, 0, 0};
#pragma unroll
        for (int k = 0; k < 4; ++k) {
            const float m2 = (k & 1) ? sq2 : cq2;
            const float m3 = (k & 2) ? sq3 : cq3;
            const float sc = m2 * m3;
            v2f Bk;
            Bk[0] = b0 * sc;
            Bk[1] = b1 * sc;
            // D += A_k (16x4 slice of M) x B_k (4x16 slice of V)
            acc = __builtin_amdgcn_wmma_f32_16x16x4_f32(
                false, A[k], false, Bk, (short)0, acc, false, false);
        }

        // <Z_q> = sum_i sign(bit_q(i)) * w_i^2 ; lane holds rows vr + 8*hi, col p
        float s[4] = {0, 0, 0, 0};
#pragma unroll
        for (int vr = 0; vr < 8; ++vr) {
            const float pw = acc[vr] * acc[vr];
            const int r = vr + (hi << 3);
#pragma unroll
            for (int q = 0; q < 4; ++q) s[q] += ((r >> q) & 1) ? -pw : pw;
        }
        // Combine lane halves: lower half needs q={0,1}, upper needs q={2,3}.
        const float t0 = hi ? s[0] : s[2];
        const float t1 = hi ? s[1] : s[3];
        const float x0 = __shfl_xor(t0, 16, 32);
        const float x1 = __shfl_xor(t1, 16, 32);
        const float oa = s[2 * hi]     + x0;
        const float ob = s[2 * hi + 1] + x1;

        // Output: float4 at (l*NB + b); each lane half writes its float2.
        const int l = ho * WO + wo;
        float2* dst = (float2*)(out + (((size_t)l * NB + b) << 2) + 2 * hi);
        *dst = make_float2(oa, ob);
    }
}

extern "C" void kernel_launch(void* const* d_in, const int* in_sizes, int n_in,
                              void* d_out, int out_size, void* d_ws, size_t ws_size,
                              hipStream_t stream) {
    const float* x     = (const float*)d_in[0];  // [32,1,512,512] float32
    const float* theta = (const float*)d_in[1];  // [4] float32
    float* out         = (float*)d_out;          // 32*4*256*256 float32

    // waves = NB * (HO/NITER) * (WO/16) = 32*32*16 = 16384 ; 8 waves/block
    const int blocks = 16384 / 8;
    quanv_wmma_kernel<<<blocks, 256, 0, stream>>>(x, theta, out);
}